// Graph2Grid_34815004901542
// MI455X (gfx1250) — compile-verified
//
#include <hip/hip_runtime.h>

typedef float v2f __attribute__((ext_vector_type(2)));
typedef float v8f __attribute__((ext_vector_type(8)));

namespace {
constexpr int Bb = 2, Tt = 12, Nn = 10000, Cg = 16;
constexpr int Cc = 8, Hh = 128, Ww = 128, HW = Hh * Ww;
constexpr int Ff = 64, Fh = 32;
constexpr int NTOT = Bb * Tt * HW;                 // 393216 elems per BN channel
constexpr size_t INTERP_ELEMS = (size_t)Bb * Tt * Cg * HW;  // 6,291,456 floats
}

// ---------------------------------------------------------------- scatter-add
__global__ void __launch_bounds__(256)
scatter_kernel(const float* __restrict__ graph, const float* __restrict__ latlon,
               float* __restrict__ interp) {
    int i = blockIdx.x * blockDim.x + threadIdx.x;
    const int total = Bb * Tt * Nn;
    if (i >= total) return;
    int n  = i % Nn;
    int bt = i / Nn;          // b*T + t
    int b  = bt / Tt;
    float2 ll = ((const float2*)latlon)[b * Nn + n];
    int lat = (int)(ll.x * (float)Hh); lat = min(max(lat, 0), Hh - 1);
    int lon = (int)(ll.y * (float)Ww); lon = min(max(lon, 0), Ww - 1);
    int pos = lat * Ww + lon;
    const float* src = graph + ((size_t)bt * Nn + n) * Cg;
    float* dst = interp + (size_t)bt * Cg * HW + pos;
#pragma unroll
    for (int c = 0; c < Cg; ++c)
        unsafeAtomicAdd(dst + (size_t)c * HW, src[c]);
}

// ------------------------------------------- fused WMMA linear + concat + stats
// Block = 256 thr = 8 waves. Each wave: 16 consecutive positions x all 64 chans.
// WMMA f32 16x16x4 layouts (ISA 7.12.2):
//   A lane l: M=l%16, a.x holds K=kbase+2*(l>=16), a.y holds K+1
//   B vgpr j, lane l: K=kbase+j+2*(l>=16), N=l%16
//   D vgpr v, lane l: M=v+8*(l>=16), N=l%16
__global__ void __launch_bounds__(256)
fused_linear_kernel(const float* __restrict__ interp, const float* __restrict__ grid,
                    const float* __restrict__ Wg, const float* __restrict__ bg,
                    const float* __restrict__ Wr, const float* __restrict__ br,
                    float* __restrict__ out, float* __restrict__ gsum,
                    float* __restrict__ gssq) {
    __shared__ float xl[Ff * 128];
    __shared__ float ssum[Ff];
    __shared__ float sssq[Ff];

    const int tid  = threadIdx.x;
    const int bt   = blockIdx.x >> 7;     // / 128 tiles per (b,t) plane
    const int tile = blockIdx.x & 127;
    const int p0   = tile * 128;

    if (tid < Ff) { ssum[tid] = 0.f; sssq[tid] = 0.f; }
    __syncthreads();

    const int w    = tid >> 5;
    const int lane = tid & 31;
    const int m    = lane & 15;
    const int hi   = lane >> 4;
    const int koff = hi << 1;
    const int pw   = p0 + w * 16;

    const float* ibase = interp + (size_t)bt * Cg * HW + pw + m;
    const float* gbase = grid   + (size_t)bt * Cc * HW + pw + m;

    v8f acc[4] = {};   // f[0:16), f[16:32) from graph; f[32:48), f[48:64) from grid

#pragma unroll
    for (int k = 0; k < 4; ++k) {          // graph half: K = 16
        const int krow = k * 4 + koff;
        const float* ap = ibase + (size_t)krow * HW;
        v2f a, b0, b1;
        a.x  = ap[0];                 a.y  = ap[HW];
        b0.x = Wg[(krow + 0) * Fh + m];        b0.y = Wg[(krow + 1) * Fh + m];
        b1.x = Wg[(krow + 0) * Fh + 16 + m];   b1.y = Wg[(krow + 1) * Fh + 16 + m];
        acc[0] = __builtin_amdgcn_wmma_f32_16x16x4_f32(false, a, false, b0,
                                                       (short)0, acc[0], false, false);
        acc[1] = __builtin_amdgcn_wmma_f32_16x16x4_f32(false, a, false, b1,
                                                       (short)0, acc[1], false, false);
    }
#pragma unroll
    for (int k = 0; k < 2; ++k) {          // grid half: K = 8
        const int krow = k * 4 + koff;
        const float* ap = gbase + (size_t)krow * HW;
        v2f a, b0, b1;
        a.x  = ap[0];                 a.y  = ap[HW];
        b0.x = Wr[(krow + 0) * Fh + m];        b0.y = Wr[(krow + 1) * Fh + m];
        b1.x = Wr[(krow + 0) * Fh + 16 + m];   b1.y = Wr[(krow + 1) * Fh + 16 + m];
        acc[2] = __builtin_amdgcn_wmma_f32_16x16x4_f32(false, a, false, b0,
                                                       (short)0, acc[2], false, false);
        acc[3] = __builtin_amdgcn_wmma_f32_16x16x4_f32(false, a, false, b1,
                                                       (short)0, acc[3], false, false);
    }

    // bias + LDS stage + per-channel partial stats
#pragma unroll
    for (int t = 0; t < 4; ++t) {
        const int f    = t * 16 + m;
        const float bias = (t < 2) ? bg[f] : br[f - 32];
        float s = 0.f, q = 0.f;
#pragma unroll
        for (int v = 0; v < 8; ++v) {
            float val = acc[t][v] + bias;
            xl[f * 128 + w * 16 + v + hi * 8] = val;
            s += val; q += val * val;
        }
        s += __shfl_down(s, 16);
        q += __shfl_down(q, 16);
        if (hi == 0) { atomicAdd(&ssum[f], s); atomicAdd(&sssq[f], q); }
    }
    __syncthreads();

    // coalesced store: 128 consecutive positions per channel row
    const int half = tid >> 7;            // 0 -> f 0..31, 1 -> f 32..63
    const int pl   = tid & 127;
    float* obase = out + (size_t)bt * Ff * HW + p0 + pl;
#pragma unroll 4
    for (int fi = 0; fi < 32; ++fi) {
        const int f = half * 32 + fi;
        obase[(size_t)f * HW] = xl[f * 128 + pl];
    }
    if (tid < Ff) {
        unsafeAtomicAdd(&gsum[tid], ssum[tid]);
        unsafeAtomicAdd(&gssq[tid], sssq[tid]);
    }
}

// ---------------------------------------------------------------- BN finalize
__global__ void finalize_stats(const float* __restrict__ gsum, const float* __restrict__ gssq,
                               const float* __restrict__ gamma, const float* __restrict__ beta,
                               float* __restrict__ scale, float* __restrict__ shift) {
    int f = threadIdx.x;
    if (f >= Ff) return;
    const float invN = 1.0f / (float)NTOT;
    float mean = gsum[f] * invN;
    float var  = gssq[f] * invN - mean * mean;
    float inv  = rsqrtf(var + 1e-5f);
    float sc   = gamma[f] * inv;
    scale[f] = sc;
    shift[f] = beta[f] - mean * sc;
}

// ---------------------------------------------------------------- normalize
__global__ void __launch_bounds__(256)
normalize_kernel(float* __restrict__ out, const float* __restrict__ scale,
                 const float* __restrict__ shift, int n4) {
    int i = blockIdx.x * blockDim.x + threadIdx.x;
    if (i >= n4) return;
    int f = (i >> 12) & 63;               // (i*4 / HW) % F ; HW/4 = 4096
    float sc = scale[f], sh = shift[f];
    float4* p = (float4*)out;
    float4 v = p[i];
    v.x = fmaf(v.x, sc, sh); v.y = fmaf(v.y, sc, sh);
    v.z = fmaf(v.z, sc, sh); v.w = fmaf(v.w, sc, sh);
    p[i] = v;
}

extern "C" void kernel_launch(void* const* d_in, const int* in_sizes, int n_in,
                              void* d_out, int out_size, void* d_ws, size_t ws_size,
                              hipStream_t stream) {
    const float* graph  = (const float*)d_in[0];   // [B,T,N,Cg]
    const float* grid   = (const float*)d_in[1];   // [B,T,C,H,W]
    const float* latlon = (const float*)d_in[2];   // [B,N,2]
    // d_in[3], d_in[4]: time indices -- unused by reference
    const float* Wg    = (const float*)d_in[5];    // [Cg, F/2]
    const float* bg    = (const float*)d_in[6];    // [F/2]
    const float* Wr    = (const float*)d_in[7];    // [C, F/2]
    const float* br    = (const float*)d_in[8];    // [F/2]
    const float* gamma = (const float*)d_in[9];    // [F]
    const float* beta  = (const float*)d_in[10];   // [F]

    float* ws     = (float*)d_ws;
    float* interp = ws;                       // 6,291,456 floats (25 MB)
    float* gsum   = ws + INTERP_ELEMS;        // 64
    float* gssq   = gsum + 64;                // 64
    float* scale  = gsum + 128;               // 64
    float* shift  = gsum + 192;               // 64
    (void)ws_size; (void)in_sizes; (void)n_in;

    // zero interp + stat accumulators (graph-capture safe)
    hipMemsetAsync(d_ws, 0, (INTERP_ELEMS + 128) * sizeof(float), stream);

    const int total = Bb * Tt * Nn;
    scatter_kernel<<<(total + 255) / 256, 256, 0, stream>>>(graph, latlon, interp);

    fused_linear_kernel<<<Bb * Tt * (HW / 128), 256, 0, stream>>>(
        interp, grid, Wg, bg, Wr, br, (float*)d_out, gsum, gssq);

    finalize_stats<<<1, 64, 0, stream>>>(gsum, gssq, gamma, beta, scale, shift);

    const int n4 = out_size / 4;
    normalize_kernel<<<(n4 + 255) / 256, 256, 0, stream>>>((float*)d_out, scale, shift, n4);
}